// HarrisNet_39505109189241
// MI455X (gfx1250) — compile-verified
//
#include <hip/hip_runtime.h>
#include <math.h>

typedef float v8f __attribute__((ext_vector_type(8)));
typedef float v2f __attribute__((ext_vector_type(2)));

static constexpr int IMG_H = 2048;
static constexpr int IMG_W = 2048;
static constexpr int NIMG  = 4;
static constexpr unsigned MED_RANK = 2097151u; // (H*W - 1) / 2

struct MedState { unsigned prefix; unsigned rank; float median; };

// band weight: g1s[t] if 0<=t<7 else 0  (g1s lives in LDS, dynamic index)
__device__ __forceinline__ float bandw(const float* g1s, int t) {
    bool ok = (unsigned)t < 7u;
    float v = g1s[ok ? t : 0];
    return ok ? v : 0.f;
}

// ---------------------------------------------------------------------------
// Kernel A: fused Sobel + products + separable 7x7 Gaussian (WMMA f32) + R
// One wave (32 threads) per 16x16 output tile. All WMMA operand paths are
// branch-free: LDS tiles are zero-padded so no EXEC-divergent ds ops remain.
// ---------------------------------------------------------------------------
__global__ __launch_bounds__(32)
void harris_response(const float* __restrict__ x, const float* __restrict__ gk,
                     float* __restrict__ R) {
    const int lane = threadIdx.x;
    const int c0 = blockIdx.x * 16;
    const int r0 = blockIdx.y * 16;
    const int img = blockIdx.z;
    const float* __restrict__ xin = x + (size_t)img * IMG_H * IMG_W;

    __shared__ float patch[24 * 24];      // input + halo 4 (1 sobel + 3 gauss)
    __shared__ float prods[3][24 * 32];   // 22x22 valid, zero-padded to 24x32
    __shared__ float vbuf[16 * 32];       // 16x22 valid, pad cols self-zero
    __shared__ float g1s[8];              // separable 1D gaussian factor

    // 1) load 24x24 patch; clamp address, select value (no divergent vmem)
    for (int idx = lane; idx < 24 * 24; idx += 32) {
        int pi = idx / 24, pj = idx % 24;
        int gr = r0 - 4 + pi, gc = c0 - 4 + pj;
        bool ok = (gr >= 0 && gr < IMG_H && gc >= 0 && gc < IMG_W);
        int ar = ok ? gr : 0, ac = ok ? gc : 0;
        float v = xin[(size_t)ar * IMG_W + ac];
        patch[idx] = ok ? v : 0.f;
    }
    // 2) 1D gaussian factor = row sums of normalized 2D kernel
    if (lane < 7) {
        float s = 0.f;
        #pragma unroll
        for (int j = 0; j < 7; j++) s += gk[lane * 7 + j];
        g1s[lane] = s;
    }
    __syncthreads();

    // 3) Sobel (cross-correlation) + products into zero-padded 24x32 tiles
    for (int idx = lane; idx < 24 * 32; idx += 32) {
        int pi = idx >> 5, pj = idx & 31;
        int gr = r0 - 3 + pi, gc = c0 - 3 + pj;
        float ixx = 0.f, iyy = 0.f, ixy = 0.f;
        if (pi < 22 && pj < 22 &&
            gr >= 0 && gr < IMG_H && gc >= 0 && gc < IMG_W) {
            const float* p = &patch[pi * 24 + pj];
            float a00 = p[0],  a01 = p[1],  a02 = p[2];
            float a10 = p[24],               a12 = p[26];
            float a20 = p[48], a21 = p[49], a22 = p[50];
            float ix = (a02 - a00) + 2.f * (a12 - a10) + (a22 - a20);
            float iy = (a20 - a00) + 2.f * (a21 - a01) + (a22 - a02);
            ixx = ix * ix; iyy = iy * iy; ixy = ix * iy;
        }
        prods[0][idx] = ixx;
        prods[1][idx] = iyy;
        prods[2][idx] = ixy;
    }

    const int half = lane >> 4;   // 0: lanes 0-15, 1: lanes 16-31
    const int n16  = lane & 15;

    // Banded weights g[k - (lane&15)] serve as BOTH vertical-pass A rows and
    // horizontal-pass B rows; compute once per lane into registers.
    float wx[6], wy[6];
    __syncthreads();
    #pragma unroll
    for (int kc = 0; kc < 6; kc++) {
        const int kb = kc * 4 + half * 2;
        wx[kc] = bandw(g1s, kb - n16);
        wy[kc] = bandw(g1s, kb + 1 - n16);
    }

    v8f S[3];
    #pragma unroll
    for (int ch = 0; ch < 3; ch++) {
        // ---- vertical pass: D = Wv(16x24-banded) x P(24x{0..15 | 16..31})
        v8f d1 = {}; v8f d2 = {};
        #pragma unroll
        for (int kc = 0; kc < 6; kc++) {
            const int kb = kc * 4 + half * 2;   // this lane's K pair (<= 23)
            v2f a, b1, b2;
            a.x = wx[kc]; a.y = wy[kc];
            b1.x = prods[ch][kb * 32 + n16];
            b1.y = prods[ch][(kb + 1) * 32 + n16];
            b2.x = prods[ch][kb * 32 + 16 + n16];
            b2.y = prods[ch][(kb + 1) * 32 + 16 + n16];
            d1 = __builtin_amdgcn_wmma_f32_16x16x4_f32(false, a, false, b1,
                                                       (short)0, d1, false, false);
            d2 = __builtin_amdgcn_wmma_f32_16x16x4_f32(false, a, false, b2,
                                                       (short)0, d2, false, false);
        }
        __syncthreads();   // prior channel's horizontal reads of vbuf done
        #pragma unroll
        for (int v = 0; v < 8; v++) {
            const int m = v + half * 8;
            vbuf[m * 32 + n16] = d1[v];
            vbuf[m * 32 + 16 + n16] = d2[v];   // cols 22..31 receive exact 0
        }
        __syncthreads();

        // ---- horizontal pass: S(16x16) = V(16x24) x Wh(24x16-banded)
        v8f s = {};
        #pragma unroll
        for (int kc = 0; kc < 6; kc++) {
            const int kb = kc * 4 + half * 2;
            v2f a, b;
            a.x = vbuf[n16 * 32 + kb];
            a.y = vbuf[n16 * 32 + kb + 1];
            b.x = wx[kc]; b.y = wy[kc];
            s = __builtin_amdgcn_wmma_f32_16x16x4_f32(false, a, false, b,
                                                      (short)0, s, false, false);
        }
        S[ch] = s;
    }

    // 4) Harris response, store R
    float* __restrict__ Ro = R + (size_t)img * IMG_H * IMG_W;
    #pragma unroll
    for (int v = 0; v < 8; v++) {
        float sxx = S[0][v], syy = S[1][v], sxy = S[2][v];
        float tr = sxx + syy;
        float r = sxx * syy - sxy * sxy - 0.05f * tr * tr;
        const int m = v + half * 8;
        Ro[(size_t)(r0 + m) * IMG_W + (c0 + n16)] = r;
    }
}

// ---------------------------------------------------------------------------
// Median via exact 3-level radix selection on order-preserving key bits
// ---------------------------------------------------------------------------
__global__ void init_state(unsigned* __restrict__ hist, MedState* __restrict__ st) {
    int i = blockIdx.x * blockDim.x + threadIdx.x;
    if (i < NIMG * 2048) hist[i] = 0u;
    if (i < NIMG) { st[i].prefix = 0u; st[i].rank = MED_RANK; st[i].median = 0.f; }
}

__global__ __launch_bounds__(256)
void hist_pass(const float* __restrict__ R, unsigned* __restrict__ hist,
               const MedState* __restrict__ st, int level) {
    __shared__ unsigned lh[2048];
    const int img = blockIdx.x >> 8;          // 256 blocks per image
    const int blk = blockIdx.x & 255;
    const int nbins = (level == 2) ? 1024 : 2048;
    for (int i = threadIdx.x; i < 2048; i += 256) lh[i] = 0u;
    __syncthreads();

    const unsigned prefix = st[img].prefix;
    const float* __restrict__ Ri = R + (size_t)img * IMG_H * IMG_W;
    const int base = blk * 16384 + threadIdx.x;   // 16384 elements per block
    for (int it = 0; it < 64; it++) {
        float f = Ri[base + it * 256];
        unsigned u = __float_as_uint(f);
        unsigned key = (u & 0x80000000u) ? ~u : (u | 0x80000000u);
        bool ok; unsigned bin;
        if (level == 0)      { ok = true;                       bin = key >> 21; }
        else if (level == 1) { ok = (key >> 21) == prefix;      bin = (key >> 10) & 2047u; }
        else                 { ok = (key >> 10) == prefix;      bin = key & 1023u; }
        if (ok) atomicAdd(&lh[bin], 1u);
    }
    __syncthreads();
    unsigned* gh = hist + img * 2048;
    for (int i = threadIdx.x; i < nbins; i += 256)
        if (lh[i]) atomicAdd(&gh[i], lh[i]);
}

__global__ void scan_pass(unsigned* __restrict__ hist, MedState* __restrict__ st, int level) {
    const int img = blockIdx.x;
    unsigned* gh = hist + img * 2048;
    const int nbins = (level == 2) ? 1024 : 2048;
    if (threadIdx.x == 0) {
        unsigned rank = st[img].rank;
        unsigned cum = 0; int b = 0;
        for (b = 0; b < nbins; b++) {
            unsigned c = gh[b];
            if (cum + c > rank) break;
            cum += c;
        }
        if (b >= nbins) b = nbins - 1;
        st[img].rank = rank - cum;
        if (level == 0)      st[img].prefix = (unsigned)b;
        else if (level == 1) st[img].prefix = (st[img].prefix << 11) | (unsigned)b;
        else {
            unsigned key = (st[img].prefix << 10) | (unsigned)b;
            unsigned u = (key & 0x80000000u) ? (key ^ 0x80000000u) : ~key;
            st[img].median = __uint_as_float(u);
        }
    }
    __syncthreads();
    for (int i = threadIdx.x; i < 2048; i += blockDim.x) gh[i] = 0u;  // next level
}

// ---------------------------------------------------------------------------
// Kernel C: threshold + separable 7x7 window max + NMS
// ---------------------------------------------------------------------------
__global__ __launch_bounds__(256)
void nms_kernel(const float* __restrict__ R, const MedState* __restrict__ st,
                float* __restrict__ out) {
    const int img = blockIdx.z;
    const int c0 = blockIdx.x * 32;
    const int r0 = blockIdx.y * 32;
    __shared__ float rt[38 * 38];
    __shared__ float hb[38 * 32];
    const float med = st[img].median;
    const float* __restrict__ Ri = R + (size_t)img * IMG_H * IMG_W;
    const int tid = threadIdx.x;

    for (int idx = tid; idx < 38 * 38; idx += 256) {
        int i = idx / 38, j = idx % 38;
        int gr = r0 - 3 + i, gc = c0 - 3 + j;
        bool ok = (gr >= 0 && gr < IMG_H && gc >= 0 && gc < IMG_W);
        int ar = ok ? gr : 0, ac = ok ? gc : 0;
        float f = Ri[(size_t)ar * IMG_W + ac];
        float thr = (f >= med) ? f : 0.f;              // Rt
        rt[idx] = ok ? thr : -INFINITY;                // -inf window padding
    }
    __syncthreads();
    for (int idx = tid; idx < 38 * 32; idx += 256) {
        int i = idx / 32, j = idx % 32;
        float m = rt[i * 38 + j];
        #pragma unroll
        for (int t = 1; t < 7; t++) m = fmaxf(m, rt[i * 38 + j + t]);
        hb[i * 32 + j] = m;
    }
    __syncthreads();
    for (int idx = tid; idx < 32 * 32; idx += 256) {
        int i = idx / 32, j = idx % 32;
        float m = hb[i * 32 + j];
        #pragma unroll
        for (int t = 1; t < 7; t++) m = fmaxf(m, hb[(i + t) * 32 + j]);
        float c = rt[(i + 3) * 38 + (j + 3)];
        out[(size_t)img * IMG_H * IMG_W + (size_t)(r0 + i) * IMG_W + (c0 + j)]
            = (c == m) ? c : 0.f;
    }
}

// ---------------------------------------------------------------------------
extern "C" void kernel_launch(void* const* d_in, const int* in_sizes, int n_in,
                              void* d_out, int out_size, void* d_ws, size_t ws_size,
                              hipStream_t stream) {
    const float* x  = (const float*)d_in[0];   // (4,1,2048,2048) f32
    const float* gk = (const float*)d_in[2];   // 7x7 gaussian f32
    float* out = (float*)d_out;

    char* ws = (char*)d_ws;
    float*    Rbuf = (float*)ws;                                   // 64 MB
    unsigned* hist = (unsigned*)(ws + (size_t)64 * 1024 * 1024);   // 32 KB
    MedState* st   = (MedState*)(ws + (size_t)64 * 1024 * 1024 + 32768);

    // 1) fused response (WMMA gaussian)
    dim3 gA(IMG_W / 16, IMG_H / 16, NIMG);
    harris_response<<<gA, dim3(32), 0, stream>>>(x, gk, Rbuf);

    // 2) exact median per image: 3-level radix selection
    init_state<<<(NIMG * 2048 + 255) / 256, 256, 0, stream>>>(hist, st);
    for (int level = 0; level < 3; level++) {
        hist_pass<<<NIMG * 256, 256, 0, stream>>>(Rbuf, hist, st, level);
        scan_pass<<<NIMG, 256, 0, stream>>>(hist, st, level);
    }

    // 3) threshold + 7x7 max + NMS
    dim3 gC(IMG_W / 32, IMG_H / 32, NIMG);
    nms_kernel<<<gC, dim3(256), 0, stream>>>(Rbuf, st, out);
}